// SogamosoGCN_7988639170621
// MI455X (gfx1250) — compile-verified
//
#include <hip/hip_runtime.h>
#include <hip/hip_bf16.h>

typedef __attribute__((ext_vector_type(2)))  float    v2f;
typedef __attribute__((ext_vector_type(8)))  float    v8f;
typedef __attribute__((ext_vector_type(16))) _Float16 v16h;

#define THREADS 256

// ---------- init: deg=1 (self loop), scalar agg=0, agg2_last=0 ----------
__global__ void k_init(float* __restrict__ deg, float* __restrict__ aggs,
                       float* __restrict__ agg2, int n) {
  int i = blockIdx.x * blockDim.x + threadIdx.x;
  if (i < n) { deg[i] = 1.0f; aggs[i] = 0.0f; }
  if (i < 8) agg2[i] = 0.0f;
}

// ---------- degree accumulation over dst ----------
__global__ void k_degree(const long long* __restrict__ dst,
                         float* __restrict__ deg, long long E) {
  long long i  = (long long)blockIdx.x * blockDim.x + threadIdx.x;
  long long st = (long long)gridDim.x * blockDim.x;
  for (; i < E; i += st) atomicAdd(&deg[(int)dst[i]], 1.0f);
}

// ---------- dinv = rsqrt(deg); p = x * dinv ----------
__global__ void k_dinv(const float* __restrict__ x, float* __restrict__ deg_dinv,
                       float* __restrict__ p, int n) {
  int i = blockIdx.x * blockDim.x + threadIdx.x;
  if (i >= n) return;
  float d  = deg_dinv[i];
  float di = d > 0.0f ? rsqrtf(d) : 0.0f;
  deg_dinv[i] = di;
  p[i] = x[i] * di;
}

// ---------- layer-1 scatter (rank-1 factored: scalar per edge) ----------
__global__ void k_edge1(const long long* __restrict__ src, const long long* __restrict__ dst,
                        const float* __restrict__ dinv, const float* __restrict__ p,
                        float* __restrict__ aggs, long long E) {
  long long i  = (long long)blockIdx.x * blockDim.x + threadIdx.x;
  long long st = (long long)gridDim.x * blockDim.x;
  for (; i < E; i += st) {
    int s = (int)src[i];
    int d = (int)dst[i];
    atomicAdd(&aggs[d], p[s] * dinv[d]);
  }
}

// ---------- hL1[n][j] = relu(s[n]*W1[j] + b1[j]),  s[n]=aggs[n]+p[n]*dinv[n] ----------
__global__ void k_node1(const float* __restrict__ aggs, const float* __restrict__ p,
                        const float* __restrict__ dinv, const float* __restrict__ W1,
                        const float* __restrict__ b1, float* __restrict__ hL1, int n) {
  int i = blockIdx.x * blockDim.x + threadIdx.x;
  if (i >= n) return;
  float sv = aggs[i] + p[i] * dinv[i];
  float* o = hL1 + (size_t)i * 16;
#pragma unroll
  for (int j = 0; j < 16; ++j) o[j] = fmaxf(fmaf(sv, W1[j], b1[j]), 0.0f);
}

// ---------- h2 = hL1 @ W2  via WMMA; one wave per 16-node tile ----------
// A (16x4 f32, 2 VGPRs): lanes 0-15 hold K={k0,k0+1}, lanes 16-31 K={k0+2,k0+3}, M=lane&15.
// B (4x16 f32, 2 VGPRs): mirrored, N=lane&15.  C/D reg r: row r + 8*(lane>=16), col lane&15.
__global__ void k_gemm(const float* __restrict__ hL1, const float* __restrict__ W2,
                       float* __restrict__ h2, int nTiles) {
  int tile = blockIdx.x * (blockDim.x >> 5) + (threadIdx.x >> 5);
  if (tile >= nTiles) return;            // wave-uniform branch: EXEC stays all-1s
  int lane = threadIdx.x & 31;
  int half = lane >> 4;                  // 0: lanes 0-15, 1: lanes 16-31
  int m    = lane & 15;
  int col  = m & 7;                      // always-in-range W2 column
  float cmask = (m < 8) ? 1.0f : 0.0f;   // zero-pad N 8->16 by value, not by EXEC

  // Preload B operands (loop-invariant per lane); unconditional, in-range loads.
  v2f bv[4];
#pragma unroll
  for (int kk = 0; kk < 4; ++kk) {
    int k0 = kk * 4 + half * 2;
    bv[kk].x = W2[ k0      * 8 + col] * cmask;
    bv[kk].y = W2[(k0 + 1) * 8 + col] * cmask;
  }

  const float* A = hL1 + (size_t)tile * 256 + (size_t)m * 16;  // this lane's row
  v8f acc = {};
#pragma unroll
  for (int kk = 0; kk < 4; ++kk) {
    int k0 = kk * 4 + half * 2;
    v2f a; a.x = A[k0]; a.y = A[k0 + 1];            // merges to global_load_b64
    acc = __builtin_amdgcn_wmma_f32_16x16x4_f32(false, a, false, bv[kk],
                                                (short)0, acc, false, false);
  }
  if (m < 8) {                                      // divergence only at stores
#pragma unroll
    for (int r = 0; r < 8; ++r)
      h2[((size_t)tile * 16 + r + half * 8) * 8 + col] = acc[r];
  }
}

// ---------- scalar tail (unused for N=100000, kept for generality) ----------
__global__ void k_gemm_tail(const float* __restrict__ hL1, const float* __restrict__ W2,
                            float* __restrict__ h2, int start, int n) {
  int i = start + blockIdx.x * blockDim.x + threadIdx.x;
  if (i >= n) return;
  for (int j = 0; j < 8; ++j) {
    float s = 0.0f;
    for (int k = 0; k < 16; ++k) s = fmaf(hL1[(size_t)i * 16 + k], W2[k * 8 + j], s);
    h2[(size_t)i * 8 + j] = s;
  }
}

// ---------- layer-2 scatter: only edges with dst == N-1 contribute ----------
__global__ void k_edge2(const long long* __restrict__ src, const long long* __restrict__ dst,
                        const float* __restrict__ dinv, const float* __restrict__ h2,
                        float* __restrict__ agg2, long long E, int last) {
  long long i  = (long long)blockIdx.x * blockDim.x + threadIdx.x;
  long long st = (long long)gridDim.x * blockDim.x;
  float dl = dinv[last];
  for (; i < E; i += st) {
    if ((int)dst[i] == last) {
      int s = (int)src[i];
      float w = dinv[s] * dl;
#pragma unroll
      for (int j = 0; j < 8; ++j) atomicAdd(&agg2[j], h2[(size_t)s * 8 + j] * w);
    }
  }
}

// ---------- self-loop + bias + relu + fc for node N-1 ----------
__global__ void k_final(const float* __restrict__ agg2, const float* __restrict__ h2,
                        const float* __restrict__ dinv, const float* __restrict__ b2,
                        const float* __restrict__ Wfc, const float* __restrict__ bfc,
                        float* __restrict__ out, int last) {
  if (threadIdx.x == 0 && blockIdx.x == 0) {
    float dl = dinv[last];
    float acc = bfc[0];
#pragma unroll
    for (int j = 0; j < 8; ++j) {
      float hf = fmaxf(agg2[j] + h2[(size_t)last * 8 + j] * dl * dl + b2[j], 0.0f);
      acc = fmaf(hf, Wfc[j], acc);
    }
    out[0] = acc;
  }
}

extern "C" void kernel_launch(void* const* d_in, const int* in_sizes, int n_in,
                              void* d_out, int out_size, void* d_ws, size_t ws_size,
                              hipStream_t stream) {
  const float*     x   = (const float*)d_in[0];
  const long long* ei  = (const long long*)d_in[1];   // int64 [2, E]
  const float*     W1  = (const float*)d_in[2];
  const float*     b1  = (const float*)d_in[3];
  const float*     W2  = (const float*)d_in[4];
  const float*     b2  = (const float*)d_in[5];
  const float*     Wfc = (const float*)d_in[6];
  const float*     bfc = (const float*)d_in[7];
  float* out = (float*)d_out;

  int       n = in_sizes[0];
  long long E = (long long)in_sizes[1] / 2;
  const long long* src = ei;
  const long long* dst = ei + E;

  // workspace layout (floats): dinv[n] | p[n] | aggs[n] | hL1[16n] | h2[8n] | agg2[8]
  float* ws   = (float*)d_ws;
  float* dinv = ws;
  float* p    = ws + (size_t)n;
  float* aggs = ws + (size_t)2 * n;
  float* hL1  = ws + (size_t)3 * n;
  float* h2   = ws + (size_t)19 * n;
  float* agg2 = ws + (size_t)27 * n;

  int nb = (n + THREADS - 1) / THREADS;
  int eb = (int)((E + THREADS - 1) / THREADS);
  if (eb > 65535) eb = 65535;   // grid-stride loops cover the rest

  k_init  <<<nb, THREADS, 0, stream>>>(dinv, aggs, agg2, n);
  k_degree<<<eb, THREADS, 0, stream>>>(dst, dinv, E);
  k_dinv  <<<nb, THREADS, 0, stream>>>(x, dinv, p, n);
  k_edge1 <<<eb, THREADS, 0, stream>>>(src, dst, dinv, p, aggs, E);
  k_node1 <<<nb, THREADS, 0, stream>>>(aggs, p, dinv, W1, b1, hL1, n);

  int fullTiles = n / 16;
  if (fullTiles > 0) {
    int gb = (fullTiles + 7) / 8;       // 8 waves (256 threads) per block
    k_gemm<<<gb, 256, 0, stream>>>(hL1, W2, h2, fullTiles);
  }
  int rem = n & 15;
  if (rem > 0)
    k_gemm_tail<<<1, 32, 0, stream>>>(hL1, W2, h2, fullTiles * 16, n);

  k_edge2 <<<eb, THREADS, 0, stream>>>(src, dst, dinv, h2, agg2, E, n - 1);
  k_final <<<1, 32, 0, stream>>>(agg2, h2, dinv, b2, Wfc, bfc, out, n - 1);
}